// Att_trans_sum_42984032698902
// MI455X (gfx1250) — compile-verified
//
#include <hip/hip_runtime.h>

// Att_trans_sum: B=16, N=256, D=64
// out = [alphas (B*N*N floats) | value (B*N*N*D floats)]  ~272.6 MB, store-BW bound.

#define BB 16
#define NN 256
#define DD 64

typedef float v2f __attribute__((ext_vector_type(2)));
typedef float v4f __attribute__((ext_vector_type(4)));
typedef float v8f __attribute__((ext_vector_type(8)));

// ---------------- Kernel 0: fold a into W -> v[d]; c1 = a.bW -----------------
__global__ void att_prep_kernel(const float* __restrict__ W,
                                const float* __restrict__ bW,
                                const float* __restrict__ a,
                                float* __restrict__ ws) {
  int d = threadIdx.x;  // 64 threads
  float acc = 0.f;
#pragma unroll 8
  for (int e = 0; e < DD; ++e) acc += a[e] * W[e * DD + d];
  ws[d] = acc;  // v[d] = sum_e a[e] * W[e,d]
  if (d == 0) {
    float c = 0.f;
    for (int e = 0; e < DD; ++e) c += a[e] * bW[e];
    ws[DD] = c;  // c1 = a . bW
  }
}

// ---------------- Kernel 1: s[row] = emb[row,:].v + c1 via fp32 WMMA ---------
// One wave per 16-row tile. A = 16x4 f32 slab of emb (A layout: lanes 0-15
// hold K=k0,k0+1; lanes 16-31 hold K=k0+2,k0+3 for row M=lane%16).
// B[k,n] = v[k0+k] broadcast across all columns N -> every C column equals
// the matvec result (robust to B's N-lane mapping; only K mapping matters).
__global__ void att_score_kernel(const float* __restrict__ emb,
                                 const float* __restrict__ ws,
                                 float* __restrict__ s_out) {
  const float* v = ws;
  const float c1 = ws[DD];
  const int lane = threadIdx.x;      // 0..31, full wave active (EXEC all 1s)
  const int half = lane >> 4;        // 0: K=k0..k0+1, 1: K=k0+2..k0+3
  const int m = lane & 15;           // row within tile
  const int base = blockIdx.x * 16;  // 256 tiles cover B*N = 4096 rows
  const float* arow = emb + (size_t)(base + m) * DD;

  v8f c = {};
#pragma unroll
  for (int k0 = 0; k0 < DD; k0 += 4) {
    const int k = k0 + 2 * half;                 // 8-byte aligned (k even)
    v2f av = *(const v2f*)(arow + k);            // A: emb[row, k..k+1]
    v2f bv = *(const v2f*)(v + k);               // B: v[k..k+1] (bcast over N)
    c = __builtin_amdgcn_wmma_f32_16x16x4_f32(
        /*neg_a=*/false, av, /*neg_b=*/false, bv,
        /*c_mod=*/(short)0, c, /*reuse_a=*/false, /*reuse_b=*/false);
  }
  // C layout: VGPR r -> (M=r, lanes 0-15) and (M=r+8, lanes 16-31); all N equal.
  if (lane == 0) {
#pragma unroll
    for (int r = 0; r < 8; ++r) s_out[base + r] = c[r] + c1;
  } else if (lane == 16) {
#pragma unroll
    for (int r = 0; r < 8; ++r) s_out[base + 8 + r] = c[r] + c1;
  }
}

// ---------------- Kernel 2: fused leaky-relu softmax + value stream ----------
// One 256-thread block per (b,i) row. Writes 1 KB alphas + 64 KB value.
__global__ void att_fused_kernel(const float* __restrict__ emb,
                                 const float* __restrict__ s,
                                 const float* __restrict__ ba,
                                 float* __restrict__ out) {
  const int tid = threadIdx.x;   // 0..255
  const int bi = blockIdx.x;     // b*N + i
  const int b = bi >> 8;

  __shared__ float red[NN];
  __shared__ float ei[DD];

  // score_ij = lrelu(s_i + s_j + ba), j = tid
  const float si = s[bi];
  const float sj = s[b * NN + tid];
  float sc = si + sj + ba[0];
  sc = sc > 0.f ? sc : 0.01f * sc;

  // max-reduce
  red[tid] = sc;
  __syncthreads();
#pragma unroll
  for (int off = 128; off > 0; off >>= 1) {
    if (tid < off) red[tid] = fmaxf(red[tid], red[tid + off]);
    __syncthreads();
  }
  const float mx = red[0];
  __syncthreads();

  // sum-reduce of exp
  const float e = __expf(sc - mx);
  red[tid] = e;
  __syncthreads();
#pragma unroll
  for (int off = 128; off > 0; off >>= 1) {
    if (tid < off) red[tid] += red[tid + off];
    __syncthreads();
  }
  const float inv = 1.f / red[0];

  // alphas[b,i,j] at out[bi*N + j]
  out[(size_t)bi * NN + tid] = e * inv;

  // stage e_i in LDS
  if (tid < DD) ei[tid] = emb[(size_t)bi * DD + tid];
  __syncthreads();

  // value[b,i,j,d] = e_i[d] * e_j[d], v4f (b128) non-temporal stores.
  // tid = jl*16 + d4; each jj-iteration writes a contiguous 4 KB chunk.
  v4f* __restrict__ vout = (v4f*)(out + (size_t)BB * NN * NN);
  const int jl = tid >> 4;
  const int d4 = tid & 15;
  const v4f ei4 = ((const v4f*)ei)[d4];  // LDS broadcast within 16-thread group
#pragma unroll 4
  for (int jj = 0; jj < 16; ++jj) {
    const int j = jj * 16 + jl;
    const v4f* ejp = (const v4f*)(emb + (size_t)(b * NN + j) * DD) + d4;
    if (jj < 15)  // prefetch next j-tile row (stays in-bounds)
      __builtin_prefetch((const void*)(ejp + (16 * DD / 4)), 0, 3);
    const v4f ej4 = *ejp;
    const v4f r = ei4 * ej4;
    __builtin_nontemporal_store(r, vout + ((size_t)bi * NN + j) * (DD / 4) + d4);
  }
}

extern "C" void kernel_launch(void* const* d_in, const int* in_sizes, int n_in,
                              void* d_out, int out_size, void* d_ws, size_t ws_size,
                              hipStream_t stream) {
  const float* emb = (const float*)d_in[0];  // [B,N,D]
  const float* W   = (const float*)d_in[1];  // [D,D]
  const float* bW  = (const float*)d_in[2];  // [D]
  const float* a   = (const float*)d_in[3];  // [D]
  const float* ba  = (const float*)d_in[4];  // [1]
  float* out = (float*)d_out;
  float* ws  = (float*)d_ws;   // [0..63]=v, [64]=c1, [128..128+4095]=s
  float* sbuf = ws + 128;

  att_prep_kernel<<<1, DD, 0, stream>>>(W, bW, a, ws);
  att_score_kernel<<<(BB * NN) / 16, 32, 0, stream>>>(emb, ws, sbuf);
  att_fused_kernel<<<BB * NN, NN, 0, stream>>>(emb, sbuf, ba, out);
}